// FP8EmbeddingBag_29386166239593
// MI455X (gfx1250) — compile-verified
//
#include <hip/hip_runtime.h>
#include <hip/hip_bf16.h>

#ifndef __has_builtin
#define __has_builtin(x) 0
#endif

#if defined(__HIP_DEVICE_COMPILE__) && \
    __has_builtin(__builtin_amdgcn_global_load_async_to_lds_b128) && \
    __has_builtin(__builtin_amdgcn_s_wait_asynccnt)
#define USE_ASYNC_GATHER 1
#else
#define USE_ASYNC_GATHER 0
#endif

#define NROWS   4096
#define BAGLEN  200
#define DIM     128
#define NWEIGHT 100000
#define NFACT   10

#if USE_ASYNC_GATHER
// Builtin signature (from hipcc diagnostic): param0 = int4 addrspace(1)*,
// param1 = int4 addrspace(3)*, then imm offset, imm cpol.
typedef int v4i __attribute__((vector_size(4 * sizeof(int))));
typedef __attribute__((address_space(1))) v4i* global_v4i_ptr;
typedef __attribute__((address_space(3))) v4i* lds_v4i_ptr;
#endif

// ---------------------------------------------------------------------------
// Exact fp8 e4m3fn quant-dequant with round-to-nearest-even.
// Valid for a >= 0, a <= 448 (guaranteed: a = x/(base*f), base = rowmax/448,
// f >= 1).  Denormal range (e < -6) handled by clamping the exponent; all
// scalings are powers of two (exact), v_rndne_f32 gives RNE.
// ---------------------------------------------------------------------------
__device__ __forceinline__ float qdq_e4m3(float a) {
    unsigned int b = __float_as_uint(a);
    int ex = (int)((b >> 23) & 0xFF) - 127;
    ex = ex < -6 ? -6 : ex;
    float step     = __builtin_ldexpf(1.0f, ex - 3);
    float inv_step = __builtin_ldexpf(1.0f, 3 - ex);
    return __builtin_rintf(a * inv_step) * step;
}

__device__ __forceinline__ float factor_val(int i) {
    // match numpy/jax arange(1.0, 2.0, 0.1) evaluated in fp64, cast to fp32
    return (float)(1.0 + 0.1 * (double)i);
}

__device__ __forceinline__ int qdq_index(float xf, float s) {
    float dq = qdq_e4m3(xf / s) * s;
    int id = (int)dq;                       // f32 -> s32 truncates toward zero
    id = id < 0 ? 0 : id;
    id = id > (NWEIGHT - 1) ? (NWEIGHT - 1) : id;
    return id;
}

// ---------------------------------------------------------------------------
// Kernel A: per row, compute base = max(|x|)/448 and, for each of the 10
// candidate factors, the row's sum of squared qdq error.  Fixed-order LDS
// tree reductions -> deterministic.
// ---------------------------------------------------------------------------
__global__ __launch_bounds__(256) void qdq_rowstats_kernel(
    const int* __restrict__ x, float* __restrict__ baseArr,
    float* __restrict__ rowsum) {
    __shared__ float red[256];
    const int r = blockIdx.x;
    const int t = threadIdx.x;

    float xf = 0.0f;
    if (t < BAGLEN) xf = (float)x[r * BAGLEN + t];

    // max reduction (values are non-negative)
    red[t] = xf;
    __syncthreads();
    for (int off = 128; off > 0; off >>= 1) {
        if (t < off) red[t] = fmaxf(red[t], red[t + off]);
        __syncthreads();
    }
    const float base = red[0] * (1.0f / 448.0f);
    if (t == 0) baseArr[r] = base;
    __syncthreads();

    for (int i = 0; i < NFACT; ++i) {
        const float s = base * factor_val(i);
        float e = 0.0f;
        if (t < BAGLEN) {
            float dq = qdq_e4m3(xf / s) * s;
            float d  = dq - xf;
            e = d * d;
        }
        red[t] = e;
        __syncthreads();
        for (int off = 128; off > 0; off >>= 1) {
            if (t < off) red[t] += red[t + off];
            __syncthreads();
        }
        if (t == 0) rowsum[r * NFACT + i] = red[0];
        __syncthreads();
    }
}

// ---------------------------------------------------------------------------
// Kernel B: deterministic combine of per-row error sums -> global MSE per
// factor -> first-min argmin -> best factor value.
// ---------------------------------------------------------------------------
__global__ __launch_bounds__(256) void pick_factor_kernel(
    const float* __restrict__ rowsum, float* __restrict__ bestF) {
    __shared__ float red[256];
    __shared__ float sums[NFACT];
    const int t = threadIdx.x;

    for (int i = 0; i < NFACT; ++i) {
        float local = 0.0f;
        for (int r = t; r < NROWS; r += 256) local += rowsum[r * NFACT + i];
        red[t] = local;
        __syncthreads();
        for (int off = 128; off > 0; off >>= 1) {
            if (t < off) red[t] += red[t + off];
            __syncthreads();
        }
        if (t == 0) sums[i] = red[0];
        __syncthreads();
    }
    if (t == 0) {
        int bi = 0;
        float bv = sums[0];
        for (int i = 1; i < NFACT; ++i)
            if (sums[i] < bv) { bv = sums[i]; bi = i; }  // first-min tie-break
        bestF[0] = factor_val(bi);
    }
}

// ---------------------------------------------------------------------------
// Kernel C: gather + mean.  One wave per output row.  Each gather is one
// GLOBAL_LOAD_ASYNC_TO_LDS_B128 (32 lanes x 16B = one full 512B embedding
// row), pipelined 6 deep on ASYNCcnt; accumulate via ds_load_b128.
// Table (51 MB) is L2-resident on MI455X (192 MB L2).
// ---------------------------------------------------------------------------
__global__ __launch_bounds__(256) void bag_gather_kernel(
    const int* __restrict__ x, const float* __restrict__ weight,
    const float* __restrict__ baseArr, const float* __restrict__ bestF,
    float* __restrict__ out) {
    __shared__ __align__(16) float slots[8][8][DIM];  // [wave][slot][floats] 32 KB
    __shared__ int idxs[8][BAGLEN];                   // 6.4 KB

    const int w    = threadIdx.x >> 5;
    const int lane = threadIdx.x & 31;
    const int r    = blockIdx.x * 8 + w;

    const float s = baseArr[r] * bestF[0];

    // Stage the 200 qdq'd indices for this row (in-wave, DS ops are in-order)
    for (int k = 0; k < 7; ++k) {
        int l = k * 32 + lane;
        if (l < BAGLEN) {
            float xf = (float)x[r * BAGLEN + l];
            idxs[w][l] = qdq_index(xf, s);
        }
    }

    float ax = 0.0f, ay = 0.0f, az = 0.0f, aw = 0.0f;

#if USE_ASYNC_GATHER
    for (int l = 0; l < BAGLEN; ++l) {
        const int id = idxs[w][l];
        const float* src = weight + (size_t)id * DIM + lane * 4;
        float* dst = &slots[w][l & 7][lane * 4];
        __builtin_amdgcn_global_load_async_to_lds_b128(
            (global_v4i_ptr)src, (lds_v4i_ptr)dst, 0, 0);
        if (l >= 5) {
            __builtin_amdgcn_s_wait_asynccnt(5);  // loads <= l-5 complete
            const float4 v =
                *(const float4*)&slots[w][(l - 5) & 7][lane * 4];
            ax += v.x; ay += v.y; az += v.z; aw += v.w;
        }
    }
    __builtin_amdgcn_s_wait_asynccnt(0);
    for (int l = BAGLEN - 5; l < BAGLEN; ++l) {
        const float4 v = *(const float4*)&slots[w][l & 7][lane * 4];
        ax += v.x; ay += v.y; az += v.z; aw += v.w;
    }
#else
    for (int l = 0; l < BAGLEN; ++l) {
        const int id = idxs[w][l];
        const float4 v =
            *(const float4*)(weight + (size_t)id * DIM + lane * 4);
        ax += v.x; ay += v.y; az += v.z; aw += v.w;
    }
#endif

    const float m = 1.0f / (float)BAGLEN;
    float4 o;
    o.x = ax * m; o.y = ay * m; o.z = az * m; o.w = aw * m;
    *(float4*)(out + (size_t)r * DIM + lane * 4) = o;
}

// ---------------------------------------------------------------------------
extern "C" void kernel_launch(void* const* d_in, const int* in_sizes, int n_in,
                              void* d_out, int out_size, void* d_ws,
                              size_t ws_size, hipStream_t stream) {
    (void)in_sizes; (void)n_in; (void)out_size; (void)ws_size;
    const int*   x      = (const int*)d_in[0];     // [4096, 200] int32
    const float* weight = (const float*)d_in[1];   // [100000, 128] f32
    float*       out    = (float*)d_out;           // [4096, 128] f32

    float* baseArr = (float*)d_ws;                 // 4096 f32
    float* rowsum  = baseArr + NROWS;              // 4096*10 f32
    float* bestF   = rowsum + NROWS * NFACT;       // 1 f32

    qdq_rowstats_kernel<<<NROWS, 256, 0, stream>>>(x, baseArr, rowsum);
    pick_factor_kernel<<<1, 256, 0, stream>>>(rowsum, bestF);
    bag_gather_kernel<<<NROWS / 8, 256, 0, stream>>>(x, weight, baseArr, bestF,
                                                     out);
}